// LatGPCN_71030169141824
// MI455X (gfx1250) — compile-verified
//
#include <hip/hip_runtime.h>

// LatGPCN on MI455X (gfx1250, wave32).
//   x[N,512], edge[2,E] (int32), A_value[E], W1[512,128], b1[128], W2[128,64], b2[64]
//   out: log_softmax over 64 classes, [N,64] f32.
// GEMMs: V_WMMA_F32_16X16X4_F32 with W panel double-buffered in LDS via the
// gfx1250 async-to-LDS path (guarded; sync fallback). Edge phase: wave-per-edge
// with f32 hardware atomics (H/X are 51 MB each -> L2-resident on 192 MB L2).

#define NNODES 100000
#define NEDGES 3200000LL

typedef float v2f __attribute__((ext_vector_type(2)));
typedef float v8f __attribute__((ext_vector_type(8)));

#if defined(__AMDGCN__) && __has_builtin(__builtin_amdgcn_global_load_async_to_lds_b128) && __has_builtin(__builtin_amdgcn_s_wait_asynccnt)
#define USE_ASYNC_LDS 1
// Builtin prototype (from hipcc diagnostic): first param is a pointer to a
// 16-byte int vector in the global (AS1) address space; dst is LDS (AS3).
typedef int v4i_vec __attribute__((vector_size(16)));
typedef __attribute__((address_space(1))) v4i_vec g_v4i;
typedef __attribute__((address_space(3))) v4i_vec l_v4i;
#else
#define USE_ASYNC_LDS 0
#endif

__device__ __forceinline__ float wave_sum(float v) {
#pragma unroll
  for (int off = 16; off >= 1; off >>= 1) v += __shfl_xor(v, off, 32);
  return v;
}
__device__ __forceinline__ float wave_max(float v) {
#pragma unroll
  for (int off = 16; off >= 1; off >>= 1) v = fmaxf(v, __shfl_xor(v, off, 32));
  return v;
}
__device__ __forceinline__ void atomic_add_f32(float* p, float v) {
  __hip_atomic_fetch_add(p, v, __ATOMIC_RELAXED, __HIP_MEMORY_SCOPE_AGENT);
}

// ---------------------------------------------------------------- zero fill
__global__ void zero_kernel(float* __restrict__ p, long long n) {
  long long i = (long long)blockIdx.x * blockDim.x + threadIdx.x;
  long long stride = (long long)gridDim.x * blockDim.x;
  for (; i < n; i += stride) p[i] = 0.f;
}

// ------------------------------------------------- X = Xin @ W + b  (WMMA f32)
// Block = 256 threads (8 waves). Block computes 128 rows x 64 cols.
// W panel chunk (32 x 64 f32 = 8 KB) staged in LDS, double-buffered; filled with
// async-to-LDS loads when available. Each wave: 16x64 tile, 4 accumulators.
#define KC 32  // k-chunk

__device__ __forceinline__ void fill_w_chunk(const float* __restrict__ W,
                                             float* lds_buf, int kb, int D,
                                             int col0, int tid) {
#pragma unroll
  for (int i = 0; i < 2; ++i) {
    int f = tid + 256 * i;            // float4 index, 0..511
    int row = f >> 4;                 // 0..31
    int c4 = f & 15;                  // 0..15 -> cols 4*c4..4*c4+3
    const float* g = W + (long long)(kb + row) * D + col0 + 4 * c4;
#if USE_ASYNC_LDS
    __builtin_amdgcn_global_load_async_to_lds_b128(
        (g_v4i*)g, (l_v4i*)(lds_buf + 4 * f), 0, 0);
#else
    float4 v = *(const float4*)g;
    *(float4*)(lds_buf + 4 * f) = v;
#endif
  }
}

__device__ __forceinline__ void wait_fill() {
#if USE_ASYNC_LDS
  __builtin_amdgcn_s_wait_asynccnt(0);
#endif
  __syncthreads();
}

__global__ __launch_bounds__(256) void gemm_bias_wmma(
    const float* __restrict__ A, const float* __restrict__ W,
    const float* __restrict__ bias, float* __restrict__ out,
    int n, int K, int D) {
  __shared__ float ldsW[2][KC * 64];

  int tid = threadIdx.x;
  int lane = tid & 31;
  int w = tid >> 5;                       // wave 0..7
  int colBlocks = D >> 6;
  int rowChunk = blockIdx.x / colBlocks;  // 128-row chunk
  int colBlock = blockIdx.x % colBlocks;
  int col0 = colBlock << 6;

  int row0 = rowChunk * 128 + w * 16;
  bool valid = row0 < n;
  int row0c = valid ? row0 : (n - 16);    // clamp; keep loads in-bounds, all
                                          // threads participate in barriers.
  int m = lane & 15;
  int h = lane >> 4;
  const float* arow = A + (long long)(row0c + m) * K;

  v8f acc[4];
#pragma unroll
  for (int t = 0; t < 4; ++t) acc[t] = (v8f){0.f,0.f,0.f,0.f,0.f,0.f,0.f,0.f};

  // prologue: fill chunk 0
  fill_w_chunk(W, &ldsW[0][0], 0, D, col0, tid);
  wait_fill();

  int nchunks = K / KC;
  for (int c = 0; c < nchunks; ++c) {
    int kb = c * KC;
    if (c + 1 < nchunks)
      fill_w_chunk(W, &ldsW[(c + 1) & 1][0], kb + KC, D, col0, tid);

    const float* Wb = &ldsW[c & 1][0];

    // preload A fragments for the whole chunk (8 x b64, one wait)
    v2f areg[8];
#pragma unroll
    for (int j = 0; j < 8; ++j) {
      areg[j].x = arow[kb + 4 * j + 2 * h];
      areg[j].y = arow[kb + 4 * j + 2 * h + 1];
    }
#pragma unroll
    for (int j = 0; j < 8; ++j) {
      int kk = 4 * j;
#pragma unroll
      for (int t = 0; t < 4; ++t) {
        v2f b;
        b.x = Wb[(kk + 2 * h) * 64 + 16 * t + m];
        b.y = Wb[(kk + 2 * h + 1) * 64 + 16 * t + m];
        acc[t] = __builtin_amdgcn_wmma_f32_16x16x4_f32(
            false, areg[j], false, b, (short)0, acc[t], false, false);
      }
    }
    wait_fill();
  }

  if (valid) {
#pragma unroll
    for (int t = 0; t < 4; ++t) {
      int col = col0 + 16 * t + m;
      float bv = bias[col];
#pragma unroll
      for (int r = 0; r < 8; ++r) {
        int row = row0 + 8 * h + r;
        out[(long long)row * D + col] = acc[t][r] + bv;
      }
    }
  }
}

// ------------------------------------------------- row squared-norms, wave/row
__global__ __launch_bounds__(256) void rownorm2_kernel(
    const float* __restrict__ H, float* __restrict__ out, int n, int D) {
  int wave = (blockIdx.x * blockDim.x + threadIdx.x) >> 5;
  int lane = threadIdx.x & 31;
  if (wave >= n) return;
  const float* row = H + (long long)wave * D;
  float s;
  if (D == 128) {
    float4 v = ((const float4*)row)[lane];
    s = v.x * v.x + v.y * v.y + v.z * v.z + v.w * v.w;
  } else {
    float2 v = ((const float2*)row)[lane];
    s = v.x * v.x + v.y * v.y;
  }
  s = wave_sum(s);
  if (lane == 0) out[wave] = s;
}

// ------------- per-edge score: S = relu(A - lh*(||Hd||^2 - 2 Hd.Xs + ||Xs||^2))
__global__ __launch_bounds__(256) void edge_score_kernel(
    const int* __restrict__ edge, const float* __restrict__ Aval,
    const float* __restrict__ H, const float* __restrict__ X,
    const float* __restrict__ HH, const float* __restrict__ XX,
    float* __restrict__ S, float* __restrict__ deg,
    long long E, int D, float lh) {
  long long e = (long long)blockIdx.x * (blockDim.x >> 5) + (threadIdx.x >> 5);
  if (e >= E) return;
  int lane = threadIdx.x & 31;
  long long dst = edge[e];
  long long src = edge[E + e];
  const float* hrow = H + dst * D;
  const float* xrow = X + src * D;
  float dot;
  if (D == 128) {
    float4 hv = ((const float4*)hrow)[lane];
    float4 xv = ((const float4*)xrow)[lane];
    dot = hv.x * xv.x + hv.y * xv.y + hv.z * xv.z + hv.w * xv.w;
  } else {
    float2 hv = ((const float2*)hrow)[lane];
    float2 xv = ((const float2*)xrow)[lane];
    dot = hv.x * xv.x + hv.y * xv.y;
  }
  dot = wave_sum(dot);
  if (lane == 0) {
    float B = HH[dst] - 2.f * dot + XX[src];
    float s = Aval[e] - lh * B;
    s = s > 0.f ? s : 0.f;
    S[e] = s;
    if (s > 0.f) atomic_add_f32(&deg[dst], s);
  }
}

// ------------- scatter: Hnew[dst,:] += (S/deg[dst]) * X[src,:]  (wave per edge)
__global__ __launch_bounds__(256) void edge_spmm_kernel(
    const int* __restrict__ edge, const float* __restrict__ S,
    const float* __restrict__ deg, const float* __restrict__ X,
    float* __restrict__ Hnew, long long E, int D) {
  long long e = (long long)blockIdx.x * (blockDim.x >> 5) + (threadIdx.x >> 5);
  if (e >= E) return;
  int lane = threadIdx.x & 31;
  float s = S[e];
  if (s <= 0.f) return;  // uniform across wave
  long long dst = edge[e];
  long long src = edge[E + e];
  float v = s / deg[dst];
  const float* xrow = X + src * D;
  float* hrow = Hnew + dst * D;
  if (D == 128) {
    float4 xv = ((const float4*)xrow)[lane];
    int base = lane << 2;
    atomic_add_f32(&hrow[base + 0], v * xv.x);
    atomic_add_f32(&hrow[base + 1], v * xv.y);
    atomic_add_f32(&hrow[base + 2], v * xv.z);
    atomic_add_f32(&hrow[base + 3], v * xv.w);
  } else {
    float2 xv = ((const float2*)xrow)[lane];
    int base = lane << 1;
    atomic_add_f32(&hrow[base + 0], v * xv.x);
    atomic_add_f32(&hrow[base + 1], v * xv.y);
  }
}

// ------------------------------------------------- (H + GAMMA*X)/(1+GAMMA), relu
__global__ __launch_bounds__(256) void combine_kernel(
    const float* __restrict__ H, const float* __restrict__ X,
    float* __restrict__ out, long long total, int relu) {
  long long i = (long long)blockIdx.x * blockDim.x + threadIdx.x;
  if (i >= total) return;
  float v = 0.5f * (H[i] + X[i]);  // GAMMA == 1
  if (relu) v = v > 0.f ? v : 0.f;
  out[i] = v;
}

// ------------------------------------------------- log_softmax over 64 classes
__global__ __launch_bounds__(256) void log_softmax64_kernel(
    const float* __restrict__ in, float* __restrict__ out, int n) {
  int wave = (blockIdx.x * blockDim.x + threadIdx.x) >> 5;
  int lane = threadIdx.x & 31;
  if (wave >= n) return;
  const float* row = in + (long long)wave * 64;
  float2 v = ((const float2*)row)[lane];
  float mx = wave_max(fmaxf(v.x, v.y));
  float se = wave_sum(__expf(v.x - mx) + __expf(v.y - mx));
  float lse = mx + __logf(se);
  float2 o;
  o.x = v.x - lse;
  o.y = v.y - lse;
  ((float2*)(out + (long long)wave * 64))[lane] = o;
}

// =====================================================================
static void run_layer(const float* Xin, int K, int D,
                      const float* W, const float* bias,
                      const int* edge, const float* Aval, float lh,
                      float* Xbuf, float* HA, float* HB,
                      float* Sbuf, float* XXbuf, float* HHbuf, float* degbuf,
                      float* outbuf, int relu, hipStream_t stream) {
  const int Nn = NNODES;
  const long long E = NEDGES;
  // GEMM: X = Xin@W + b ; block covers 128 rows x 64 cols
  {
    int colBlocks = D >> 6;
    int rowChunks = (Nn + 127) / 128;
    gemm_bias_wmma<<<rowChunks * colBlocks, 256, 0, stream>>>(
        Xin, W, bias, Xbuf, Nn, K, D);
  }
  int nodeWaveBlocks = (Nn + 7) / 8;
  rownorm2_kernel<<<nodeWaveBlocks, 256, 0, stream>>>(Xbuf, XXbuf, Nn, D);

  int edgeBlocks = (int)((E + 7) / 8);
  long long total = (long long)Nn * D;
  int zeroHBlocks = (int)((total + 255) / 256);
  int zeroNBlocks = (Nn + 255) / 256;

  const float* Hcur = Xbuf;
  float* bufs[2] = {HA, HB};
  for (int it = 0; it < 4; ++it) {
    float* Hnext = bufs[it & 1];
    zero_kernel<<<zeroNBlocks, 256, 0, stream>>>(degbuf, Nn);
    zero_kernel<<<zeroHBlocks, 256, 0, stream>>>(Hnext, total);
    rownorm2_kernel<<<nodeWaveBlocks, 256, 0, stream>>>(Hcur, HHbuf, Nn, D);
    edge_score_kernel<<<edgeBlocks, 256, 0, stream>>>(
        edge, Aval, Hcur, Xbuf, HHbuf, XXbuf, Sbuf, degbuf, E, D, lh);
    edge_spmm_kernel<<<edgeBlocks, 256, 0, stream>>>(
        edge, Sbuf, degbuf, Xbuf, Hnext, E, D);
    Hcur = Hnext;
  }
  combine_kernel<<<zeroHBlocks, 256, 0, stream>>>(Hcur, Xbuf, outbuf, total, relu);
}

extern "C" void kernel_launch(void* const* d_in, const int* in_sizes, int n_in,
                              void* d_out, int out_size, void* d_ws, size_t ws_size,
                              hipStream_t stream) {
  (void)in_sizes; (void)n_in; (void)out_size; (void)ws_size;
  const float* x    = (const float*)d_in[0];
  const int*   edge = (const int*)d_in[1];   // [2,E] int32
  const float* Aval = (const float*)d_in[2];
  const float* W1   = (const float*)d_in[3];
  const float* b1   = (const float*)d_in[4];
  const float* W2   = (const float*)d_in[5];
  const float* b2   = (const float*)d_in[6];

  const long long Nn = NNODES;
  float* ws = (float*)d_ws;
  float* X1 = ws;                 // N*128
  float* HA = X1 + Nn * 128;      // N*128
  float* HB = HA + Nn * 128;      // N*128
  float* H1 = HB + Nn * 128;      // N*128 (layer-1 activated output)
  float* S  = H1 + Nn * 128;      // E
  float* XX = S + NEDGES;         // N
  float* HH = XX + Nn;            // N
  float* dg = HH + Nn;            // N

  // Layer 1: K=512 -> D=128, lamba_half = 0.5, relu after combine
  run_layer(x, 512, 128, W1, b1, edge, Aval, 0.5f,
            X1, HA, HB, S, XX, HH, dg, H1, /*relu=*/1, stream);

  // Layer 2: K=128 -> D=64. Reuse X1 as X2, HA/HB as ping-pong.
  // After 4 iters Hcur = HB, Xbuf = X1, so HA is free for the combine output.
  run_layer(H1, 128, 64, W2, b2, edge, Aval, 0.5f,
            X1, HA, HB, S, XX, HH, dg, HA, /*relu=*/0, stream);

  // log_softmax over 64 classes -> d_out
  log_softmax64_kernel<<<(NNODES + 7) / 8, 256, 0, stream>>>(
      HA, (float*)d_out, NNODES);
}